// ResGCN_58317065945292
// MI455X (gfx1250) — compile-verified
//
#include <hip/hip_runtime.h>
#include <math.h>

typedef __attribute__((ext_vector_type(2))) float v2f;
typedef __attribute__((ext_vector_type(8))) float v8f;

// ---------------- fill / degree / norm ----------------

__global__ void k_fill0(float* __restrict__ p, int n) {
  int i = blockIdx.x * blockDim.x + threadIdx.x;
  if (i < n) p[i] = 0.0f;
}

__global__ void k_fill1(float* __restrict__ p, int n) {  // deg init = self-loop weight 1
  int i = blockIdx.x * blockDim.x + threadIdx.x;
  if (i < n) p[i] = 1.0f;
}

__global__ void k_deg_accum(const int* __restrict__ dst, const float* __restrict__ w,
                            float* __restrict__ deg, int E) {
  int e = blockIdx.x * blockDim.x + threadIdx.x;
  if (e < E) unsafeAtomicAdd(&deg[dst[e]], w[e]);
}

__global__ void k_dinv(float* __restrict__ deg, int n) {
  int i = blockIdx.x * blockDim.x + threadIdx.x;
  if (i < n) { float d = deg[i]; deg[i] = d > 0.0f ? rsqrtf(d) : 0.0f; }
}

__global__ void k_norm(const int* __restrict__ src, const int* __restrict__ dst,
                       const float* __restrict__ w, const float* __restrict__ dinv,
                       float* __restrict__ nrm, int E, int n) {
  int e = blockIdx.x * blockDim.x + threadIdx.x;
  if (e >= E + n) return;
  if (e < E) nrm[e] = dinv[src[e]] * w[e] * dinv[dst[e]];
  else       { float di = dinv[e - E]; nrm[e] = di * di; }
}

// ---------------- WMMA f32 GEMM: Y[N,128] = X[N,128] @ W[128,128] ----------------
// 16 rows per block; 8 waves, each wave owns one 16-col tile; 32 x WMMA_F32_16X16X4_F32.
// A 16x4 f32 layout: lanes 0-15 M=0..15 hold K=k0,k0+1; lanes 16-31 hold K=k0+2,k0+3.
// B 4x16: VGPR0 = rows K=k0 (lanes 0-15) / K=k0+2 (lanes 16-31), VGPR1 = K+1.
// C/D: VGPR r -> (M=r, N=lane) lanes 0-15 ; (M=r+8) lanes 16-31.
__global__ void k_gemm128(const float* __restrict__ X, const float* __restrict__ W,
                          float* __restrict__ Y, int nrows) {
  const int lane = threadIdx.x & 31;
  const int wave = threadIdx.x >> 5;        // column tile 0..7
  const int rowBase = blockIdx.x << 4;
  if (rowBase >= nrows) return;
  const int colBase = wave << 4;
  const int m = lane & 15;
  const int half = lane >> 4;               // 0 or 1
  const float* xrow = X + (size_t)(rowBase + m) * 128;

  v8f c = {};
  #pragma unroll
  for (int k0 = 0; k0 < 128; k0 += 4) {
    const int ka = k0 + 2 * half;
    v2f a; a.x = xrow[ka];                       a.y = xrow[ka + 1];
    v2f b; b.x = W[(size_t)ka * 128 + colBase + m];
           b.y = W[(size_t)(ka + 1) * 128 + colBase + m];
    c = __builtin_amdgcn_wmma_f32_16x16x4_f32(false, a, false, b, (short)0, c,
                                              false, false);
  }
  #pragma unroll
  for (int r = 0; r < 8; ++r) {
    int row = rowBase + r + 8 * half;
    Y[(size_t)row * 128 + colBase + m] = c[r];
  }
}

// ---------------- scatter (segment-sum): out[dst] += norm * xw[src], 128 feats ----------------
__global__ void k_scatter128(const float* __restrict__ xw, const int* __restrict__ src,
                             const int* __restrict__ dst, const float* __restrict__ nrm,
                             float* __restrict__ out, int E, int n) {
  int e = blockIdx.x * (blockDim.x >> 5) + (threadIdx.x >> 5);  // one wave per edge
  int lane = threadIdx.x & 31;
  if (e >= E + n) return;
  int s, d;
  if (e < E) { s = src[e]; d = dst[e]; } else { s = d = e - E; }
  const float r = nrm[e];
  const float4 v = *(const float4*)(xw + (size_t)s * 128 + lane * 4);
  float* o = out + (size_t)d * 128 + lane * 4;
  unsafeAtomicAdd(o + 0, v.x * r);
  unsafeAtomicAdd(o + 1, v.y * r);
  unsafeAtomicAdd(o + 2, v.z * r);
  unsafeAtomicAdd(o + 3, v.w * r);
}

__global__ void k_relubias(float* __restrict__ h, const float* __restrict__ b, int total) {
  int i = blockIdx.x * blockDim.x + threadIdx.x;
  if (i < total) h[i] = fmaxf(h[i] + b[i & 127], 0.0f);
}

// ---------------- h = (l2norm(h) + x) * 0.5, wave per node ----------------
__global__ void k_resnorm(float* __restrict__ h, const float* __restrict__ x, int n) {
  int idx = blockIdx.x * (blockDim.x >> 5) + (threadIdx.x >> 5);
  int lane = threadIdx.x & 31;
  if (idx >= n) return;
  float* hp = h + (size_t)idx * 128 + lane * 4;
  float4 v = *(float4*)hp;
  float s = v.x * v.x + v.y * v.y + v.z * v.z + v.w * v.w;
  #pragma unroll
  for (int off = 16; off > 0; off >>= 1) s += __shfl_xor(s, off, 32);
  float inv = 1.0f / fmaxf(sqrtf(s), 1e-12f);
  const float4 xv = *(const float4*)(x + (size_t)idx * 128 + lane * 4);
  v.x = (v.x * inv + xv.x) * 0.5f;
  v.y = (v.y * inv + xv.y) * 0.5f;
  v.z = (v.z * inv + xv.z) * 0.5f;
  v.w = (v.w * inv + xv.w) * 0.5f;
  *(float4*)hp = v;
}

// ---------------- Y[N,4] = H[N,128] @ Wf[128,4], wave per node ----------------
__global__ void k_gemm4(const float* __restrict__ H, const float* __restrict__ Wf,
                        float* __restrict__ Y, int n) {
  int idx = blockIdx.x * (blockDim.x >> 5) + (threadIdx.x >> 5);
  int lane = threadIdx.x & 31;
  if (idx >= n) return;
  const float4 hv = *(const float4*)(H + (size_t)idx * 128 + lane * 4);
  float a0 = 0, a1 = 0, a2 = 0, a3 = 0;
  const float hvv[4] = {hv.x, hv.y, hv.z, hv.w};
  #pragma unroll
  for (int j = 0; j < 4; ++j) {
    int k = lane * 4 + j;
    a0 += hvv[j] * Wf[k * 4 + 0];
    a1 += hvv[j] * Wf[k * 4 + 1];
    a2 += hvv[j] * Wf[k * 4 + 2];
    a3 += hvv[j] * Wf[k * 4 + 3];
  }
  #pragma unroll
  for (int off = 16; off > 0; off >>= 1) {
    a0 += __shfl_xor(a0, off, 32);
    a1 += __shfl_xor(a1, off, 32);
    a2 += __shfl_xor(a2, off, 32);
    a3 += __shfl_xor(a3, off, 32);
  }
  if (lane == 0) {
    float4 o; o.x = a0; o.y = a1; o.z = a2; o.w = a3;
    *(float4*)(Y + (size_t)idx * 4) = o;
  }
}

// ---------------- scatter 4-wide: hf[dst] += norm * x4[src] ----------------
__global__ void k_scatter4(const float* __restrict__ x4, const int* __restrict__ src,
                           const int* __restrict__ dst, const float* __restrict__ nrm,
                           float* __restrict__ out, int E, int n) {
  int e = blockIdx.x * blockDim.x + threadIdx.x;
  if (e >= E + n) return;
  int s, d;
  if (e < E) { s = src[e]; d = dst[e]; } else { s = d = e - E; }
  const float r = nrm[e];
  const float4 v = *(const float4*)(x4 + (size_t)s * 4);
  float* o = out + (size_t)d * 4;
  unsafeAtomicAdd(o + 0, v.x * r);
  unsafeAtomicAdd(o + 1, v.y * r);
  unsafeAtomicAdd(o + 2, v.z * r);
  unsafeAtomicAdd(o + 3, v.w * r);
}

// ---------------- predictor: gather 2x4 + bias, 8x4 matmul, log_softmax ----------------
__global__ void k_pred(const float* __restrict__ hf, const int* __restrict__ pe,
                       const float* __restrict__ Wp, const float* __restrict__ bp,
                       const float* __restrict__ bf, float* __restrict__ out, int EP) {
  int p = blockIdx.x * blockDim.x + threadIdx.x;
  if (p >= EP) return;
  int u = pe[2 * p], v = pe[2 * p + 1];
  const float4 hu = *(const float4*)(hf + (size_t)u * 4);
  const float4 hv = *(const float4*)(hf + (size_t)v * 4);
  float hc[8];
  hc[0] = hu.x + bf[0]; hc[1] = hu.y + bf[1]; hc[2] = hu.z + bf[2]; hc[3] = hu.w + bf[3];
  hc[4] = hv.x + bf[0]; hc[5] = hv.y + bf[1]; hc[6] = hv.z + bf[2]; hc[7] = hv.w + bf[3];
  float lg[4];
  #pragma unroll
  for (int c = 0; c < 4; ++c) {
    float a = bp[c];
    #pragma unroll
    for (int k = 0; k < 8; ++k) a += hc[k] * Wp[k * 4 + c];
    lg[c] = a;
  }
  float m = fmaxf(fmaxf(lg[0], lg[1]), fmaxf(lg[2], lg[3]));
  float sum = expf(lg[0] - m) + expf(lg[1] - m) + expf(lg[2] - m) + expf(lg[3] - m);
  float lse = m + logf(sum);
  float4 o;
  o.x = lg[0] - lse; o.y = lg[1] - lse; o.z = lg[2] - lse; o.w = lg[3] - lse;
  *(float4*)(out + (size_t)p * 4) = o;
}

// ---------------- host ----------------

extern "C" void kernel_launch(void* const* d_in, const int* in_sizes, int n_in,
                              void* d_out, int out_size, void* d_ws, size_t ws_size,
                              hipStream_t stream) {
  const float* x  = (const float*)d_in[0];
  const int*   ei = (const int*)  d_in[1];
  const float* ew = (const float*)d_in[2];
  const int*   pe = (const int*)  d_in[3];
  const float* W1 = (const float*)d_in[4];
  const float* b1 = (const float*)d_in[5];
  const float* W2 = (const float*)d_in[6];
  const float* b2 = (const float*)d_in[7];
  const float* W3 = (const float*)d_in[8];
  const float* b3 = (const float*)d_in[9];
  const float* Wf = (const float*)d_in[10];
  const float* bf = (const float*)d_in[11];
  const float* Wp = (const float*)d_in[12];
  const float* bp = (const float*)d_in[13];

  const int D   = 128;
  const int N   = in_sizes[0] / D;
  const int E   = in_sizes[1] / 2;
  const int EP  = in_sizes[3] / 2;
  const int tot = E + N;
  const int* src = ei;
  const int* dst = ei + E;

  float* ws = (float*)d_ws;
  size_t o = 0;
  float* bufG = ws + o; o += (size_t)N * D;   // GEMM output
  float* bufH = ws + o; o += (size_t)N * D;   // conv output / h
  float* dinv = ws + o; o += (size_t)N;       // deg -> dinv in place
  float* nrm  = ws + o; o += (size_t)tot;     // per-edge norm
  float* out4 = ws + o; o += (size_t)N * 4;   // h @ Wf
  float* hf   = ws + o; o += (size_t)N * 4;   // final conv output

  const int B = 256;

  // degree + normalization (graph-only; shared by all 4 convs)
  k_fill1<<<(N + B - 1) / B, B, 0, stream>>>(dinv, N);
  k_deg_accum<<<(E + B - 1) / B, B, 0, stream>>>(dst, ew, dinv, E);
  k_dinv<<<(N + B - 1) / B, B, 0, stream>>>(dinv, N);
  k_norm<<<(tot + B - 1) / B, B, 0, stream>>>(src, dst, ew, dinv, nrm, E, N);

  const float* hin  = x;
  const float* Wl[3] = {W1, W2, W3};
  const float* bl[3] = {b1, b2, b3};
  for (int l = 0; l < 3; ++l) {
    k_gemm128<<<(N + 15) / 16, B, 0, stream>>>(hin, Wl[l], bufG, N);
    k_fill0<<<(N * D + B - 1) / B, B, 0, stream>>>(bufH, N * D);
    k_scatter128<<<(tot + 7) / 8, B, 0, stream>>>(bufG, src, dst, nrm, bufH, E, N);
    k_relubias<<<(N * D + B - 1) / B, B, 0, stream>>>(bufH, bl[l], N * D);
    hin = bufH;
  }

  k_resnorm<<<(N + 7) / 8, B, 0, stream>>>(bufH, x, N);

  k_gemm4<<<(N + 7) / 8, B, 0, stream>>>(bufH, Wf, out4, N);
  k_fill0<<<(N * 4 + B - 1) / B, B, 0, stream>>>(hf, N * 4);
  k_scatter4<<<(tot + B - 1) / B, B, 0, stream>>>(out4, src, dst, nrm, hf, E, N);

  k_pred<<<(EP + B - 1) / B, B, 0, stream>>>(hf, pe, Wp, bp, bf, (float*)d_out, EP);
}